// ResidualGCN_16329465660184
// MI455X (gfx1250) — compile-verified
//
#include <hip/hip_runtime.h>
#include <stdint.h>

// ---------------- problem constants (match reference) ----------------
#define NN 100000
#define EE 1600000
#define HH 128        // hidden = feature dim
#define LL 8
#define GG 256
#define BN_EPS 1e-5f

#define USE_TDM 1     // stage W into LDS via Tensor Data Mover (tensor_load_to_lds)

typedef __attribute__((ext_vector_type(2))) float    v2f;
typedef __attribute__((ext_vector_type(8))) float    v8f;
typedef __attribute__((ext_vector_type(4))) uint32_t u32x4;
typedef __attribute__((ext_vector_type(8))) uint32_t u32x8;

// ---------------- small utility kernels ----------------
__global__ void zero_f(float* __restrict__ p, long n) {
    long i = (long)blockIdx.x * blockDim.x + threadIdx.x;
    if (i < n) p[i] = 0.0f;
}
__global__ void fill_f(float* __restrict__ p, long n, float v) {
    long i = (long)blockIdx.x * blockDim.x + threadIdx.x;
    if (i < n) p[i] = v;
}

// ---------------- degree / norm ----------------
__global__ void deg_init(float* __restrict__ deg) {
    int i = blockIdx.x * blockDim.x + threadIdx.x;
    if (i < NN) deg[i] = 1.0f;   // self-loop: deg = in-degree + 1
}
__global__ void deg_count(const int* __restrict__ ei, float* __restrict__ deg) {
    int e = blockIdx.x * blockDim.x + threadIdx.x;
    if (e < EE) atomicAdd(&deg[ei[EE + e]], 1.0f);   // dst = ei[1][e]
}
__global__ void deg_rsqrt(float* __restrict__ deg) {
    int i = blockIdx.x * blockDim.x + threadIdx.x;
    if (i < NN) deg[i] = rsqrtf(deg[i]);             // in-place -> dinv
}

// ---------------- WMMA GEMM: C[M,128] = A[M,128] @ W[128,128] (+bias) ----------------
// One block = 8 waves = one 32-row panel. Waves 0-3 -> rows[0:16), waves 4-7 -> rows[16:32).
// Each wave owns a PAIR of 16-col tiles, so every A fragment feeds 2 WMMAs (A reuse x2).
// W (64 KB) is DMA'd into LDS by the Tensor Data Mover (wave 0 issues, TENSORcnt waits),
// then read back as conflict-free ds loads. M % 32 == 0 (100000 = 3125*32).
__global__ __launch_bounds__(256) void gemm128_wmma(
    const float* __restrict__ A, const float* __restrict__ W,
    const float* __restrict__ bias, float* __restrict__ C, int hasBias)
{
    __shared__ float Ws[HH * HH];  // 64 KB
    const int t = threadIdx.x;

#if USE_TDM
    if (t < 32) {   // wave 0 issues one TDM descriptor for the whole 64 KB tile
        const uint64_t ga = (uint64_t)(uintptr_t)W;
        // Generic LDS pointer: high 32 bits = shared aperture, low 32 bits = LDS byte offset.
        const uint32_t lds_base = (uint32_t)(uintptr_t)(void*)Ws;
        u32x4 g0;
        g0[0] = 1u;                                            // count=1 (valid), user mode
        g0[1] = lds_base;                                      // lds_addr (bytes)
        g0[2] = (uint32_t)ga;                                  // global_addr[31:0]
        g0[3] = (uint32_t)((ga >> 32) & 0x1FFFFFFu)            // global_addr[56:32]
              | (2u << 30);                                    // type = 2 ("image")
        u32x8 g1;
        g1[0] = (2u << 16);          // workgroup_mask=0, data_size=2 (4 bytes)
        g1[1] = (0x4000u << 16);     // tensor_dim0[15:0]=16384 at bits[63:48]
        g1[2] = (1u << 16);          // tensor_dim0[31:16]=0, tensor_dim1[15:0]=1
        g1[3] = (0x4000u << 16);     // tile_dim0=16384 at bits[127:112]
        g1[4] = 1u;                  // tile_dim1=1, tile_dim2=0
        g1[5] = 16384u;              // tensor_dim0_stride[31:0]
        g1[6] = 0u;                  // stride hi / tensor_dim1_stride lo
        g1[7] = 0u;
        // Pass Ws as an operand so the LDS buffer ESCAPES into the asm: the
        // "memory" clobber then forces the compiler to treat Ws as written by
        // the TDM (otherwise the ds loads below get DCE'd as reads of
        // never-written memory).
        asm volatile("tensor_load_to_lds %0, %1"
                     :
                     : "s"(g0), "s"(g1), "v"((void*)Ws)
                     : "memory");
#if defined(__has_builtin) && __has_builtin(__builtin_amdgcn_s_wait_tensorcnt)
        __builtin_amdgcn_s_wait_tensorcnt(0);
#else
        asm volatile("s_wait_tensorcnt 0x0" ::: "memory");
#endif
    }
    __syncthreads();
#else
    {   // fallback: cooperative 128-bit staged copy of W into LDS
        const float4* Wv = reinterpret_cast<const float4*>(W);
        float4*       Sv = reinterpret_cast<float4*>(Ws);
        #pragma unroll
        for (int i = t; i < HH * HH / 4; i += 256) Sv[i] = Wv[i];
    }
    __syncthreads();
#endif

    const int wave  = t >> 5;
    const int lane  = t & 31;
    const int mhalf = wave >> 2;        // 0/1 -> which 16-row tile of the 32-row panel
    const int npair = wave & 3;         // 0..3 -> 32-column pair
    const int m     = lane & 15;        // A row within tile
    const int hi    = lane >> 4;        // half-wave selects K pair
    const int koff  = hi * 2;
    const int nc0   = npair * 32 + (lane & 15);
    const int nc1   = nc0 + 16;
    const long row  = (long)blockIdx.x * 32 + mhalf * 16 + m;
    const float* Arow = A + row * HH;

    v8f acc0 = {0.f, 0.f, 0.f, 0.f, 0.f, 0.f, 0.f, 0.f};
    v8f acc1 = {0.f, 0.f, 0.f, 0.f, 0.f, 0.f, 0.f, 0.f};
    #pragma unroll 4
    for (int kk = 0; kk < HH; kk += 4) {
        v2f a, b0, b1;
        a.x  = Arow[kk + koff];
        a.y  = Arow[kk + koff + 1];
        b0.x = Ws[(kk + koff)     * HH + nc0];
        b0.y = Ws[(kk + koff + 1) * HH + nc0];
        b1.x = Ws[(kk + koff)     * HH + nc1];
        b1.y = Ws[(kk + koff + 1) * HH + nc1];
        acc0 = __builtin_amdgcn_wmma_f32_16x16x4_f32(
            false, a, false, b0, (short)0, acc0, false, false);
        acc1 = __builtin_amdgcn_wmma_f32_16x16x4_f32(
            false, a, false, b1, (short)0, acc1, false, false);
    }

    // D layout: VGPR r -> lanes 0-15: M=r, lanes 16-31: M=r+8; N = lane&15
    const long mbase = (long)blockIdx.x * 32 + mhalf * 16 + hi * 8;
    const float bc0 = hasBias ? bias[nc0] : 0.0f;
    const float bc1 = hasBias ? bias[nc1] : 0.0f;
    #pragma unroll
    for (int r = 0; r < 8; ++r) {
        C[(mbase + r) * HH + nc0] = acc0[r] + bc0;
        C[(mbase + r) * HH + nc1] = acc1[r] + bc1;
    }
}

// ---------------- edge scatter: agg[dst] += (dinv[src]*dinv[dst]) * hw[src] ----------------
// One wave per edge: 32 lanes x float4 = 128 channels. hw/agg are L2-resident (51 MB each).
__global__ __launch_bounds__(256) void edge_scatter(
    const int* __restrict__ ei, const float* __restrict__ dinv,
    const float* __restrict__ hw, float* __restrict__ agg)
{
    long tid  = (long)blockIdx.x * 256 + threadIdx.x;
    long e    = tid >> 5;
    int  lane = (int)(tid & 31);
    if (e >= EE) return;
    int src = ei[e];
    int dst = ei[EE + e];
    float nrm = dinv[src] * dinv[dst];
    float4 v = reinterpret_cast<const float4*>(hw + (long)src * HH)[lane];
    float* d = agg + (long)dst * HH + lane * 4;
    atomicAdd(d + 0, nrm * v.x);
    atomicAdd(d + 1, nrm * v.y);
    atomicAdd(d + 2, nrm * v.z);
    atomicAdd(d + 3, nrm * v.w);
}

// ---------------- combine self-loop + bias (in place) and reduce BN stats ----------------
__global__ __launch_bounds__(128) void combine_stats(
    float* __restrict__ agg, const float* __restrict__ hw,
    const float* __restrict__ dinv, const float* __restrict__ bias,
    float* __restrict__ bnsum, float* __restrict__ bnsq, int nodesPerBlock)
{
    const int c  = threadIdx.x;
    const int n0 = blockIdx.x * nodesPerBlock;
    const int n1 = min(n0 + nodesPerBlock, NN);
    const float bc = bias[c];
    float s = 0.0f, s2 = 0.0f;
    for (int n = n0; n < n1; ++n) {
        float di = dinv[n];
        long idx = (long)n * HH + c;
        float v = agg[idx] + di * di * hw[idx] + bc;
        agg[idx] = v;
        s += v;
        s2 += v * v;
    }
    atomicAdd(&bnsum[c], s);
    atomicAdd(&bnsq[c], s2);
}

// ---------------- BN apply (+optional ReLU, +optional residual) ----------------
__global__ __launch_bounds__(256) void bn_apply(
    const float* __restrict__ src, const float* __restrict__ bnsum,
    const float* __restrict__ bnsq, const float* __restrict__ g,
    const float* __restrict__ be, const float* __restrict__ res,
    float* __restrict__ dst, int doRelu)
{
    long i = (long)blockIdx.x * 256 + threadIdx.x;
    if (i >= (long)NN * HH) return;
    int c = (int)(i & (HH - 1));
    float mu  = bnsum[c] * (1.0f / NN);
    float var = bnsq[c] * (1.0f / NN) - mu * mu;   // population var, matches h.var(0)
    float v = (src[i] - mu) * rsqrtf(var + BN_EPS) * g[c] + be[c];
    if (doRelu) v = fmaxf(v, 0.0f);
    if (res) v += res[i];
    dst[i] = v;
}

// ---------------- segment readout: sum / max / count per graph ----------------
__device__ __forceinline__ void atomicMaxF(float* addr, float v) {
    int* ai = (int*)addr;
    int old = __float_as_int(*addr);
    while (__int_as_float(old) < v) {
        int assumed = old;
        old = atomicCAS(ai, assumed, __float_as_int(v));
        if (old == assumed) break;
    }
}
__global__ __launch_bounds__(256) void seg_scatter(
    const float* __restrict__ h, const int* __restrict__ batch,
    float* __restrict__ gsum, float* __restrict__ gmax, float* __restrict__ gcnt)
{
    long i = (long)blockIdx.x * 256 + threadIdx.x;
    if (i >= (long)NN * HH) return;
    int n = (int)(i >> 7);
    int c = (int)(i & (HH - 1));
    int g = batch[n];
    float v = h[i];
    atomicAdd(&gsum[(long)g * HH + c], v);
    atomicMaxF(&gmax[(long)g * HH + c], v);
    if (c == 0) atomicAdd(&gcnt[g], 1.0f);
}

// z = concat([mean, max, sum], axis=1)  -> [G, 3H]
__global__ void build_z(const float* __restrict__ gsum, const float* __restrict__ gmax,
                        const float* __restrict__ gcnt, float* __restrict__ z)
{
    int i = blockIdx.x * blockDim.x + threadIdx.x;
    if (i >= GG * 3 * HH) return;
    int g = i / (3 * HH);
    int c = i % (3 * HH);
    float v;
    if (c < HH)            v = gsum[g * HH + c] / fmaxf(gcnt[g], 1.0f);
    else if (c < 2 * HH)   v = gmax[g * HH + (c - HH)];
    else                   v = gsum[g * HH + (c - 2 * HH)];
    z[i] = v;
}

// tiny dense layer: Y[M,Nc] = X[M,K] @ W[K,Nc] + b (+optional ReLU). G=256 -> noise FLOPs.
__global__ void dense(const float* __restrict__ X, const float* __restrict__ W,
                      const float* __restrict__ b, float* __restrict__ Y,
                      int M, int K, int Nc, int doRelu)
{
    int i = blockIdx.x * blockDim.x + threadIdx.x;
    if (i >= M * Nc) return;
    int r = i / Nc, col = i % Nc;
    float acc = b[col];
    for (int k = 0; k < K; ++k) acc += X[r * K + k] * W[k * Nc + col];
    if (doRelu) acc = fmaxf(acc, 0.0f);
    Y[i] = acc;
}

// ---------------- host-side pipeline ----------------
extern "C" void kernel_launch(void* const* d_in, const int* in_sizes, int n_in,
                              void* d_out, int out_size, void* d_ws, size_t ws_size,
                              hipStream_t stream) {
    const float* x    = (const float*)d_in[0];
    const int*   ei   = (const int*)  d_in[1];
    const int*   batch= (const int*)  d_in[2];
    const float* Wp   = (const float*)d_in[3];
    const float* bp   = (const float*)d_in[4];
    const float* W1   = (const float*)d_in[5];
    const float* b1   = (const float*)d_in[6];
    const float* g1   = (const float*)d_in[7];
    const float* be1  = (const float*)d_in[8];
    const float* W2   = (const float*)d_in[9];
    const float* b2   = (const float*)d_in[10];
    const float* g2   = (const float*)d_in[11];
    const float* be2  = (const float*)d_in[12];
    const float* Wa   = (const float*)d_in[13];
    const float* ba   = (const float*)d_in[14];
    const float* Wb   = (const float*)d_in[15];
    const float* bb   = (const float*)d_in[16];
    const float* Wc   = (const float*)d_in[17];
    const float* bc   = (const float*)d_in[18];
    float* out = (float*)d_out;

    // workspace partition (~206 MB of f32)
    float* ws = (float*)d_ws;
    size_t off = 0;
    auto alloc = [&](size_t n) { float* p = ws + off; off += n; return p; };
    float* dinv  = alloc(NN);
    float* h     = alloc((size_t)NN * HH);
    float* o     = alloc((size_t)NN * HH);
    float* hw    = alloc((size_t)NN * HH);
    float* agg   = alloc((size_t)NN * HH);
    float* bnsum = alloc(HH);
    float* bnsq  = alloc(HH);       // contiguous after bnsum (zeroed together)
    float* gsum  = alloc((size_t)GG * HH);
    float* gmax  = alloc((size_t)GG * HH);
    float* gcnt  = alloc(GG);
    float* z     = alloc((size_t)GG * 3 * HH);
    float* z1    = alloc((size_t)GG * 2 * HH);
    float* z2    = alloc((size_t)GG * HH);
    (void)ws_size; (void)n_in; (void)in_sizes; (void)out_size;

    const long NH = (long)NN * HH;                   // 12,800,000
    const int  NH_BLK   = (int)(NH / 256);           // 50000 (exact)
    const int  ESC_BLK  = (int)((long)EE * 32 / 256);// 200000 (exact)
    const int  GEMM_BLK = NN / 32;                   // 3125 (exact, 32-row panels)
    const int  NPB = 64;                             // nodes per combine block
    const int  CMB_BLK = (NN + NPB - 1) / NPB;       // 1563

    // 1) degrees -> dinv
    deg_init <<<(NN + 255) / 256, 256, 0, stream>>>(dinv);
    deg_count<<<(EE + 255) / 256, 256, 0, stream>>>(ei, dinv);
    deg_rsqrt<<<(NN + 255) / 256, 256, 0, stream>>>(dinv);

    // 2) input projection: h = x @ Wp + bp
    gemm128_wmma<<<GEMM_BLK, 256, 0, stream>>>(x, Wp, bp, h, 1);

    // 3) L residual blocks, each = 2x (GCNConv -> BN), ReLU in between, residual add
    auto conv = [&](const float* hin, const float* W, const float* b,
                    const float* g, const float* be,
                    const float* res, float* dst, int doRelu) {
        gemm128_wmma<<<GEMM_BLK, 256, 0, stream>>>(hin, W, nullptr, hw, 0);
        zero_f<<<NH_BLK, 256, 0, stream>>>(agg, NH);
        edge_scatter<<<ESC_BLK, 256, 0, stream>>>(ei, dinv, hw, agg);
        zero_f<<<1, 256, 0, stream>>>(bnsum, 2 * HH);  // bnsum + bnsq
        combine_stats<<<CMB_BLK, 128, 0, stream>>>(agg, hw, dinv, b, bnsum, bnsq, NPB);
        bn_apply<<<NH_BLK, 256, 0, stream>>>(agg, bnsum, bnsq, g, be, res, dst, doRelu);
    };
    for (int l = 0; l < LL; ++l) {
        const size_t wO = (size_t)l * HH * HH, vO = (size_t)l * HH;
        // o = relu(bn(gcn(h, W1, b1)))
        conv(h, W1 + wO, b1 + vO, g1 + vO, be1 + vO, nullptr, o, 1);
        // h = h + bn(gcn(o, W2, b2))
        conv(o, W2 + wO, b2 + vO, g2 + vO, be2 + vO, h, h, 0);
    }

    // 4) readout
    zero_f<<<(GG * HH + 255) / 256, 256, 0, stream>>>(gsum, GG * HH);
    fill_f<<<(GG * HH + 255) / 256, 256, 0, stream>>>(gmax, GG * HH, -3.402823466e38f);
    zero_f<<<1, 256, 0, stream>>>(gcnt, GG);
    seg_scatter<<<NH_BLK, 256, 0, stream>>>(h, batch, gsum, gmax, gcnt);
    build_z<<<(GG * 3 * HH + 255) / 256, 256, 0, stream>>>(gsum, gmax, gcnt, z);

    // 5) readout MLP: 3H -> 2H -> H -> 1
    dense<<<(GG * 2 * HH) / 256, 256, 0, stream>>>(z,  Wa, ba, z1, GG, 3 * HH, 2 * HH, 1);
    dense<<<(GG * HH) / 256,     256, 0, stream>>>(z1, Wb, bb, z2, GG, 2 * HH, HH,     1);
    dense<<<1,                   256, 0, stream>>>(z2, Wc, bc, out, GG, HH,    1,      0);
}